// QLayer_65481071395344
// MI455X (gfx1250) — compile-verified
//
#include <hip/hip_runtime.h>
#include <hip/hip_bf16.h>

// Closed-form evaluation of the reference circuit.
//
// Derivation (exact, not an approximation):
//  * All RX/RY gates act on distinct qubits before any CNOT, so the
//    pre-entangler state is the product state  (x) U_i|0>,
//    U_i = RY(params[i]) RX(x[b,i]).
//  * |<1|U_i|0>|^2 = q_i  with  1 - 2 q_i = cos(params[i]) * cos(x[b,i]).
//  * Each CNOT(i,i+1) is a basis-state permutation (flip bit i+1 iff bit i
//    is 1); the composed inverse permutation maps bit i -> XOR of bits 0..i.
//  * Hence <Z_i> after the chain = E[(-1)^(b0^...^bi)] which factorizes over
//    the independent qubit bits:
//        out[b,i] = prod_{k=0..i} cos(params[k]) * cos(x[b,k])
//
// This turns a 2 GB / 2 GFLOP state-vector simulation into a 1024-element
// prefix product: launch-latency bound, no matrix ops remain, so WMMA is
// not applicable to the optimal algorithm for this problem.

#define N_QUBITS 16
#define BATCH    64

__global__ __launch_bounds__(BATCH)
void qlayer_closed_form(const float* __restrict__ x,       // (BATCH, N_QUBITS)
                        const float* __restrict__ params,  // (N_QUBITS)
                        float* __restrict__ out)           // (BATCH, N_QUBITS)
{
    const int b = threadIdx.x;  // one thread per batch row; blockDim.x == BATCH

    // params are wave-uniform -> compiler emits scalar loads from the
    // constant path; cos(params[k]) is computed once per lane (trivial).
    float cp[N_QUBITS];
#pragma unroll
    for (int k = 0; k < N_QUBITS; ++k) {
        cp[k] = __builtin_cosf(params[k]);
    }

    // Load this row's 16 angles as 4 x float4 (b128 loads).
    const float4* xrow = reinterpret_cast<const float4*>(x + b * N_QUBITS);
    float xv[N_QUBITS];
#pragma unroll
    for (int v = 0; v < N_QUBITS / 4; ++v) {
        float4 f = xrow[v];
        xv[4 * v + 0] = f.x;
        xv[4 * v + 1] = f.y;
        xv[4 * v + 2] = f.z;
        xv[4 * v + 3] = f.w;
    }

    // Prefix product of cos(params[k]) * cos(x[b,k]).
    float res[N_QUBITS];
    float prod = 1.0f;
#pragma unroll
    for (int k = 0; k < N_QUBITS; ++k) {
        prod *= cp[k] * __builtin_cosf(xv[k]);
        res[k] = prod;
    }

    // Store the row back as 4 x float4 (b128 stores).
    float4* orow = reinterpret_cast<float4*>(out + b * N_QUBITS);
#pragma unroll
    for (int v = 0; v < N_QUBITS / 4; ++v) {
        float4 f;
        f.x = res[4 * v + 0];
        f.y = res[4 * v + 1];
        f.z = res[4 * v + 2];
        f.w = res[4 * v + 3];
        orow[v] = f;
    }
}

extern "C" void kernel_launch(void* const* d_in, const int* in_sizes, int n_in,
                              void* d_out, int out_size, void* d_ws, size_t ws_size,
                              hipStream_t stream) {
    (void)in_sizes; (void)n_in; (void)d_ws; (void)ws_size; (void)out_size;
    const float* x      = (const float*)d_in[0];  // (64, 16) float32
    const float* params = (const float*)d_in[1];  // (16,)    float32
    float*       out    = (float*)d_out;          // (64, 16) float32

    qlayer_closed_form<<<1, BATCH, 0, stream>>>(x, params, out);
}